// CorrBlock_69853348102329
// MI455X (gfx1250) — compile-verified
//
#include <hip/hip_runtime.h>
#include <cstdint>
#include <cstddef>

#define NLVL   4
#define RAD    3
#define KOFF   49           // (2*RAD+1)^2
#define B_     2
#define H_     64
#define W_     64
#define C_     64
#define N_     4096
#define OUTD   (NLVL * KOFF * C_)   // 12544

typedef float f2 __attribute__((ext_vector_type(2)));

// pyramid layout in workspace (floats)
#define LVL0_OFF 0
#define LVL1_OFF (B_*H_*W_*C_)                    // 524288
#define LVL2_OFF (LVL1_OFF + B_*32*32*C_)         // 655360
#define LVL3_OFF (LVL2_OFF + B_*16*16*C_)         // 688128

// ---- CDNA5 async-to-LDS probe -------------------------------------------
#if defined(__AMDGCN__) && __has_builtin(__builtin_amdgcn_global_load_async_to_lds_b128) && __has_builtin(__builtin_amdgcn_s_wait_asynccnt)
#define USE_ASYNC 1
#else
#define USE_ASYNC 0
#if defined(__AMDGCN__)
#warning "gfx1250 async-to-LDS builtins unavailable on this toolchain; using direct-load fallback"
#endif
#endif

static __device__ __forceinline__ float warp_sum32(float v) {
  #pragma unroll
  for (int off = 16; off > 0; off >>= 1) v += __shfl_xor(v, off, 32);
  return v;
}

// ---------------------------------------------------------------------------
// Kernel 1: L2-normalize fmap2 along C. One wave per pixel, 2 ch/lane.
// fmap2 is read exactly once -> non-temporal loads.
__global__ __launch_bounds__(256) void norm_kernel(const float* __restrict__ fmap2,
                                                   float* __restrict__ lvl0) {
  int wave = (blockIdx.x * blockDim.x + threadIdx.x) >> 5;
  int lane = threadIdx.x & 31;
  if (wave >= B_ * H_ * W_) return;
  const f2* src = (const f2*)(fmap2 + (size_t)wave * C_);
  f2 v = __builtin_nontemporal_load(src + lane);
  float ss = warp_sum32(v.x * v.x + v.y * v.y);
  float s = 1.0f / (sqrtf(ss) + 1e-6f);
  f2 o; o.x = v.x * s; o.y = v.y * s;
  ((f2*)(lvl0 + (size_t)wave * C_))[lane] = o;   // pyramid stays cached (regular store)
}

// ---------------------------------------------------------------------------
// Kernel 2: antialiased bilinear 2x downsample (jax.image.resize semantics):
// taps {2i-1..2i+2}, weights [1,3,3,1]/8, renormalized per-dimension at edges.
__global__ __launch_bounds__(256) void down_kernel(const float* __restrict__ src,
                                                   float* __restrict__ dst,
                                                   int Hin, int Win) {
  const int Hout = Hin >> 1, Wout = Win >> 1;
  int wave = (blockIdx.x * blockDim.x + threadIdx.x) >> 5;
  int lane = threadIdx.x & 31;
  if (wave >= B_ * Hout * Wout) return;
  int x = wave % Wout;
  int y = (wave / Wout) % Hout;
  int b = wave / (Wout * Hout);

  const float base[4] = {0.125f, 0.375f, 0.375f, 0.125f};
  float wy[4], wx[4]; int iy[4], ix[4];
  float sy = 0.f, sx = 0.f;
  #pragma unroll
  for (int t = 0; t < 4; ++t) {
    iy[t] = 2 * y - 1 + t;
    wy[t] = (iy[t] >= 0 && iy[t] < Hin) ? base[t] : 0.f; sy += wy[t];
    ix[t] = 2 * x - 1 + t;
    wx[t] = (ix[t] >= 0 && ix[t] < Win) ? base[t] : 0.f; sx += wx[t];
  }
  float rs = (1.f / sy) * (1.f / sx);

  f2 acc; acc.x = 0.f; acc.y = 0.f;
  #pragma unroll
  for (int ty = 0; ty < 4; ++ty) {
    if (wy[ty] == 0.f) continue;
    #pragma unroll
    for (int tx = 0; tx < 4; ++tx) {
      if (wx[tx] == 0.f) continue;
      float w = wy[ty] * wx[tx] * rs;
      const f2* p = (const f2*)(src + (((size_t)b * Hin + iy[ty]) * Win + ix[tx]) * C_);
      f2 v = p[lane];
      acc.x += w * v.x; acc.y += w * v.y;
    }
  }
  ((f2*)(dst + (((size_t)b * Hout + y) * Wout + x) * C_))[lane] = acc;
}

// ---------------------------------------------------------------------------
// Kernel 3: per (b, n, level): stage the 8x8x64 patch around the query into
// LDS (async-to-LDS on CDNA5), then 8 waves compute the 49 bilinear samples
// from LDS and stream results out with non-temporal stores.
__global__ __launch_bounds__(256) void corr_sample_kernel(const float* __restrict__ pyr,
                                                          const float* __restrict__ coords,
                                                          float* __restrict__ out) {
  __shared__ __align__(16) float patch[8][8][C_];   // 16 KB

  const int bid   = blockIdx.x;
  const int level = bid & 3;
  const int n     = (bid >> 2) & (N_ - 1);
  const int b     = bid >> 14;
  const int tid   = threadIdx.x;
  const int wave  = tid >> 5;
  const int lane  = tid & 31;

  const int Wl = W_ >> level, Hl = H_ >> level;
  const int lvl_off[4] = {LVL0_OFF, LVL1_OFF, LVL2_OFF, LVL3_OFF};
  const float* lev = pyr + lvl_off[level];

  const float cx = coords[((size_t)b * N_ + n) * 2 + 0];
  const float cy = coords[((size_t)b * N_ + n) * 2 + 1];
  const float inv = 1.0f / (float)(1 << level);
  const float xs = cx * inv, ys = cy * inv;
  const int x0c = (int)xs;     // xs >= 0
  const int y0c = (int)ys;

  // ---- stage 8x8 patch (rows clamped to the level bounds) into LDS ----
  {
    int r  = tid >> 2;            // 0..63 : (ty,tx) cell
    int p  = tid & 3;             // 16-float chunk within the 64-ch row
    int ty = r >> 3, tx = r & 7;
    int yi = min(max(y0c - 3 + ty, 0), Hl - 1);
    int xi = min(max(x0c - 3 + tx, 0), Wl - 1);
    const float* g = lev + (((size_t)b * Hl + yi) * Wl + xi) * C_ + p * 16;
    float* l = &patch[ty][tx][p * 16];
#if USE_ASYNC
    // exact builtin pointee type: int __attribute__((vector_size(16)))
    typedef int v4i __attribute__((vector_size(16)));
    typedef __attribute__((address_space(1))) v4i v4i_g;   // prints as "__device__" in HIP
    typedef __attribute__((address_space(3))) v4i v4i_l;   // LDS
    v4i_g* gp = (v4i_g*)(uintptr_t)g;                      // AS1 addr == generic addr
    v4i_l* lp = (v4i_l*)(uint32_t)(uintptr_t)l;            // LDS offset = low 32 bits
    __builtin_amdgcn_global_load_async_to_lds_b128(gp, lp, 0,  0);
    __builtin_amdgcn_global_load_async_to_lds_b128(gp, lp, 16, 0);
    __builtin_amdgcn_global_load_async_to_lds_b128(gp, lp, 32, 0);
    __builtin_amdgcn_global_load_async_to_lds_b128(gp, lp, 48, 0);
    __builtin_amdgcn_s_wait_asynccnt(0);
#else
    const float4* gv = (const float4*)g;
    float4* lv = (float4*)l;
    #pragma unroll
    for (int q = 0; q < 4; ++q) lv[q] = gv[q];
#endif
  }
  __syncthreads();

  float* obase = out + ((size_t)b * N_ + n) * OUTD + (size_t)level * (KOFF * C_);

  // ---- 8 waves cover the 49 offsets; 2 channels per lane ----
  for (int k = wave; k < KOFF; k += 8) {
    const int dy = k / 7 - RAD;
    const int dx = k % 7 - RAD;
    float x = fminf(fmaxf(xs + (float)dx, 0.f), (float)(Wl - 1));
    float y = fminf(fmaxf(ys + (float)dy, 0.f), (float)(Hl - 1));
    int x0 = (int)x, y0 = (int)y;                 // == clipped floor
    int x1 = min(x0 + 1, Wl - 1);
    int y1 = min(y0 + 1, Hl - 1);
    // reference weights (collapse to 0 at clipped borders, as in JAX)
    float fx1 = (float)x1 - x, fx0 = x - (float)x0;
    float fy1 = (float)y1 - y, fy0 = y - (float)y0;
    float wa = fx1 * fy1, wb = fx1 * fy0, wc = fx0 * fy1, wd = fx0 * fy0;
    // map absolute indices into the staged patch
    int tx0 = min(max(x0 - x0c + 3, 0), 7);
    int tx1 = min(max(x1 - x0c + 3, 0), 7);
    int ty0 = min(max(y0 - y0c + 3, 0), 7);
    int ty1 = min(max(y1 - y0c + 3, 0), 7);

    f2 va = ((const f2*)&patch[ty0][tx0][0])[lane];
    f2 vb = ((const f2*)&patch[ty1][tx0][0])[lane];
    f2 vc = ((const f2*)&patch[ty0][tx1][0])[lane];
    f2 vd = ((const f2*)&patch[ty1][tx1][0])[lane];

    f2 o;
    o.x = wa * va.x + wb * vb.x + wc * vc.x + wd * vd.x;
    o.y = wa * va.y + wb * vb.y + wc * vc.y + wd * vd.y;
    // 411 MB output stream (> 192 MB L2): non-temporal so the pyramid stays hot
    __builtin_nontemporal_store(o, (f2*)(obase + (size_t)k * C_) + lane);
  }
}

// ---------------------------------------------------------------------------
extern "C" void kernel_launch(void* const* d_in, const int* in_sizes, int n_in,
                              void* d_out, int out_size, void* d_ws, size_t ws_size,
                              hipStream_t stream) {
  (void)in_sizes; (void)n_in; (void)out_size; (void)ws_size;
  // d_in[0] = fmap1 (unused by the reference), d_in[1] = fmap2, d_in[2] = coords
  const float* fmap2  = (const float*)d_in[1];
  const float* coords = (const float*)d_in[2];
  float* out = (float*)d_out;
  float* pyr = (float*)d_ws;     // 696320 floats (~2.79 MB)

  // normalize: one wave per pixel -> 8192 waves = 1024 blocks
  norm_kernel<<<(B_ * H_ * W_ * 32) / 256, 256, 0, stream>>>(fmap2, pyr + LVL0_OFF);
  // pyramid levels 1..3
  down_kernel<<<(B_ * 32 * 32 * 32) / 256, 256, 0, stream>>>(pyr + LVL0_OFF, pyr + LVL1_OFF, 64, 64);
  down_kernel<<<(B_ * 16 * 16 * 32) / 256, 256, 0, stream>>>(pyr + LVL1_OFF, pyr + LVL2_OFF, 32, 32);
  down_kernel<<<(B_ *  8 *  8 * 32) / 256, 256, 0, stream>>>(pyr + LVL2_OFF, pyr + LVL3_OFF, 16, 16);
  // sampling: one block per (b, n, level) = 32768 blocks
  corr_sample_kernel<<<B_ * N_ * NLVL, 256, 0, stream>>>(pyr, coords, out);
}